// MyEmbedding_73246372266075
// MI455X (gfx1250) — compile-verified
//
#include <hip/hip_runtime.h>
#include <hip/hip_bf16.h>

typedef __attribute__((ext_vector_type(2))) float v2f;
typedef __attribute__((ext_vector_type(8))) float v8f;

// ---------------------------------------------------------------------------
// GEMM 1: [image_embedding_raw ; region_emb_raw] (768 x 768) @ fc_w (768x4096)
//          + fc_b  ->  ws rows 0..767
// One wave (32 threads) computes one 16x16 tile of C using
// V_WMMA_F32_16X16X4_F32, looping K in steps of 4.
//
// VGPR layouts (ISA 7.12.2):
//   A 16x4 f32 : lanes 0-15 hold M=0..15 {K=k0,k0+1}; lanes 16-31 {K=k0+2,k0+3}
//   B 4x16 f32 : lanes 0-15 hold N=0..15 {K=k0,k0+1}; lanes 16-31 {K=k0+2,k0+3}
//   C/D 16x16  : vgpr r, lanes 0-15 -> M=r,  N=lane
//                        lanes 16-31 -> M=r+8, N=lane-16
// ---------------------------------------------------------------------------
__global__ __launch_bounds__(32) void gemm_fc_kernel(
    const float* __restrict__ img,   // (256, 768)
    const float* __restrict__ reg,   // (512, 768)
    const float* __restrict__ w,     // (768, 4096)
    const float* __restrict__ bias,  // (4096)
    float* __restrict__ out)         // ws rows 0..767, stride 4096
{
    const int lane = threadIdx.x & 31;
    const int lrow = lane & 15;
    const int half = lane >> 4;           // 0: K pair {0,1}; 1: K pair {2,3}
    const int n0   = blockIdx.x * 16;
    const int m0   = blockIdx.y * 16;

    const int m = m0 + lrow;
    const float* Arow = (m < 256) ? (img + (size_t)m * 768)
                                  : (reg + (size_t)(m - 256) * 768);
    const int col = n0 + lrow;

    v8f c = {};
#pragma unroll 4
    for (int k0 = 0; k0 < 768; k0 += 4) {
        const int k = k0 + 2 * half;
        v2f a = *(const v2f*)(Arow + k);             // 64-bit contiguous load
        v2f b;
        b.x = w[(size_t)k       * 4096 + col];
        b.y = w[(size_t)(k + 1) * 4096 + col];
        c = __builtin_amdgcn_wmma_f32_16x16x4_f32(
                /*neg_a=*/false, a, /*neg_b=*/false, b,
                /*c_mod=*/(short)0, c, /*reuse_a=*/false, /*reuse_b=*/false);
    }

    const float bv = bias[col];
#pragma unroll
    for (int r = 0; r < 8; ++r) {
        const int row = m0 + r + 8 * half;
        out[(size_t)row * 4096 + col] = c[r] + bv;
    }
}

// ---------------------------------------------------------------------------
// GEMM 2: mask_emb_raw (16 x 255) @ mask_fc_w (255x4096) + mask_fc_b
//          -> ws rows 768..783.  K padded 255 -> 256 with zeros (per-lane
//          selects, no EXEC divergence).
// ---------------------------------------------------------------------------
__global__ __launch_bounds__(32) void gemm_msk_kernel(
    const float* __restrict__ msk,   // (16, 255)
    const float* __restrict__ w,     // (255, 4096)
    const float* __restrict__ bias,  // (4096)
    float* __restrict__ out)         // ws, writes rows 768..783
{
    const int lane = threadIdx.x & 31;
    const int lrow = lane & 15;
    const int half = lane >> 4;
    const int n0   = blockIdx.x * 16;

    const float* Arow = msk + (size_t)lrow * 255;
    const int col = n0 + lrow;

    v8f c = {};
#pragma unroll 4
    for (int k0 = 0; k0 < 256; k0 += 4) {
        const int k = k0 + 2 * half;
        v2f a, b;
        a.x = (k     < 255) ? Arow[k]     : 0.0f;
        a.y = (k + 1 < 255) ? Arow[k + 1] : 0.0f;
        b.x = (k     < 255) ? w[(size_t)k       * 4096 + col] : 0.0f;
        b.y = (k + 1 < 255) ? w[(size_t)(k + 1) * 4096 + col] : 0.0f;
        c = __builtin_amdgcn_wmma_f32_16x16x4_f32(
                false, a, false, b, (short)0, c, false, false);
    }

    const float bv = bias[col];
#pragma unroll
    for (int r = 0; r < 8; ++r) {
        const int row = 768 + r + 8 * half;
        out[(size_t)row * 4096 + col] = c[r] + bv;
    }
}

// ---------------------------------------------------------------------------
// Gather: one 256-thread block per output row (B*T = 4096 rows of 4096 f32).
// Index decode is block-uniform; copies are 4 x float4 (b128) per thread.
// Table layout per batch b:
//   [0,32000)        weight
//   [32000,32002)    image_token_weight
//   [32002,32004)    region_token_weight
//   [32004,32132)    ws image rows   (b*128 + i)
//   [32132,32396)    vre: j=idx-32132, region=j/33, slot=j%33,
//                    area=region2areas[b,region];
//                    slot<32 -> ws[256 + b*256 + area*32 + slot]
//                    slot==32-> ws[768 + b*8 + area]
// ---------------------------------------------------------------------------
__global__ __launch_bounds__(256) void gather_rows_kernel(
    const float* __restrict__ weight,
    const float* __restrict__ itw,
    const float* __restrict__ rtw,
    const int*   __restrict__ text,
    const int*   __restrict__ r2a,
    const float* __restrict__ ws,
    float* __restrict__ out)
{
    const int bt  = blockIdx.x;          // 0 .. B*T-1
    const int b   = bt >> 11;            // / 2048
    const int idx = text[bt];

    const float* src;
    if (idx < 32000) {
        src = weight + (size_t)idx * 4096;
    } else if (idx < 32002) {
        src = itw + (size_t)(idx - 32000) * 4096;
    } else if (idx < 32004) {
        src = rtw + (size_t)(idx - 32002) * 4096;
    } else if (idx < 32132) {
        src = ws + (size_t)(b * 128 + (idx - 32004)) * 4096;
    } else {
        const int j      = idx - 32132;
        const int region = j / 33;
        const int slot   = j - region * 33;
        const int area   = r2a[b * 8 + region];
        if (slot < 32)
            src = ws + (size_t)(256 + b * 256 + area * 32 + slot) * 4096;
        else
            src = ws + (size_t)(768 + b * 8 + area) * 4096;
    }

    float4*       dst4 = (float4*)(out + (size_t)bt * 4096);
    const float4* src4 = (const float4*)src;
    const int tid = threadIdx.x;
#pragma unroll
    for (int i = 0; i < 4; ++i)
        dst4[tid + 256 * i] = src4[tid + 256 * i];
}

extern "C" void kernel_launch(void* const* d_in, const int* in_sizes, int n_in,
                              void* d_out, int out_size, void* d_ws, size_t ws_size,
                              hipStream_t stream) {
    const float* weight  = (const float*)d_in[0];   // (32000, 4096)
    const float* itw     = (const float*)d_in[1];   // (2, 4096)
    const float* rtw     = (const float*)d_in[2];   // (2, 4096)
    const float* img     = (const float*)d_in[3];   // (2, 128, 768)
    const float* reg     = (const float*)d_in[4];   // (2, 8, 32, 768)
    const float* mskraw  = (const float*)d_in[5];   // (2, 8, 255)
    const float* fc_w    = (const float*)d_in[6];   // (768, 4096)
    const float* fc_b    = (const float*)d_in[7];   // (4096)
    const float* mfc_w   = (const float*)d_in[8];   // (255, 4096)
    const float* mfc_b   = (const float*)d_in[9];   // (4096)
    const int*   text    = (const int*)d_in[10];    // (2, 2048)
    const int*   r2a     = (const int*)d_in[11];    // (2, 8)
    float*       out     = (float*)d_out;           // (2, 2048, 4096)
    float*       ws      = (float*)d_ws;            // needs 784*4096*4 B ~ 12.8 MB

    // Projections into workspace (WMMA f32 16x16x4).
    gemm_fc_kernel<<<dim3(4096 / 16, 768 / 16), 32, 0, stream>>>(
        img, reg, fc_w, fc_b, ws);
    gemm_msk_kernel<<<dim3(4096 / 16, 1), 32, 0, stream>>>(
        mskraw, mfc_w, mfc_b, ws);

    // Row gather (memory-bound part; fully vectorized b128 copies).
    gather_rows_kernel<<<2 * 2048, 256, 0, stream>>>(
        weight, itw, rtw, text, r2a, ws, out);
}